// discrete_29678224015561
// MI455X (gfx1250) — compile-verified
//
#include <hip/hip_runtime.h>

namespace {

constexpr int NG = 128;   // grid points per axis
constexpr int NP = 129;   // padded: one high-side ghost layer (low ghost is never reachable)
constexpr long long PADN = (long long)NP * NP * NP;

// ---- shared math: Min & Gibou (2007) non-oscillatory quadratic interpolation ----

__device__ __forceinline__ float mg_eval(const float nb[4][4][4],
                                         float xd, float yd, float zd)
{
  // min |second difference| along each axis over the 2x2x2 core (8 candidates each).
  // (a - 2*b) + c: 2*b is FP-exact, so this bit-matches the reference's a - 2.0*b + c.
  float mnx = 3.402823466e+38f, mny = mnx, mnz = mnx;
#pragma unroll
  for (int b = 1; b < 3; ++b) {
#pragma unroll
    for (int c = 1; c < 3; ++c) {
#pragma unroll
      for (int p = 0; p < 2; ++p) {
        float dxx = (nb[p + 2][b][c] - 2.0f * nb[p + 1][b][c]) + nb[p][b][c];
        float dyy = (nb[b][p + 2][c] - 2.0f * nb[b][p + 1][c]) + nb[b][p][c];
        float dzz = (nb[b][c][p + 2] - 2.0f * nb[b][c][p + 1]) + nb[b][c][p];
        mnx = fminf(mnx, fabsf(dxx));
        mny = fminf(mny, fabsf(dyy));
        mnz = fminf(mnz, fabsf(dzz));
      }
    }
  }
  // trilinear blend of the 2x2x2 core
  float c00 = nb[1][1][1] + zd * (nb[1][1][2] - nb[1][1][1]);
  float c01 = nb[1][2][1] + zd * (nb[1][2][2] - nb[1][2][1]);
  float c10 = nb[2][1][1] + zd * (nb[2][1][2] - nb[2][1][1]);
  float c11 = nb[2][2][1] + zd * (nb[2][2][2] - nb[2][2][1]);
  float c0  = c00 + yd * (c01 - c00);
  float c1  = c10 + yd * (c11 - c10);
  float clin = c0 + xd * (c1 - c0);
  return clin - 0.5f * (mnx * (xd * (1.0f - xd)) +
                        mny * (yd * (1.0f - yd)) +
                        mnz * (zd * (1.0f - zd)));
}

struct Idx { int i; float d; };

// Matches reference: i = clip(int((p - xg[0]) / dx), 2, 128) with xg[0] = -dx,
// d = (p - xg[i]) / dx.  For p in [0,1) the clip at 128 never binds, so we
// clamp to 127 (keeps ghost handling to a single high-side layer).
__device__ __forceinline__ Idx axis_idx(float p)
{
  const float dx = 1.0f / 127.0f;
  float t = (p + dx) / dx;
  int i = (int)t;
  i = i < 2 ? 2 : (i > 127 ? 127 : i);
  float d = (p - (float)(i - 1) * dx) / dx;
  return Idx{i, d};
}

// ---- prep: build ghost-padded 129^3 copies of both grids into workspace ----
// Padded value at a ghost coordinate reproduces the reference's sequential
// extrapolation ext_z(ext_y(ext_x(G))) exactly (x2 is FP-exact, so the
// contraction-free composition bit-matches).
__global__ __launch_bounds__(256) void pad_ghost_kernel(
    const float* __restrict__ gm, const float* __restrict__ gp,
    float* __restrict__ pad)
{
  int li = blockIdx.x * 256 + threadIdx.x;
  if (li >= (int)PADN) return;
  const float* g = blockIdx.y ? gp : gm;
  float* dst = pad + (size_t)blockIdx.y * (size_t)PADN;
  int z  = li % NP;
  int yx = li / NP;
  int y  = yx % NP;
  int x  = yx / NP;
  float v;
  if (x < NG && y < NG && z < NG) {
    v = g[(x << 14) + (y << 7) + z];
  } else {
    auto G = [&](int a, int b, int c) { return g[(a << 14) + (b << 7) + c]; };
    auto R = [&](int b, int c) {
      return (x < NG) ? G(x, b, c) : (2.0f * G(127, b, c) - G(126, b, c));
    };
    auto Q = [&](int c) {
      return (y < NG) ? R(y, c) : (2.0f * R(127, c) - R(126, c));
    };
    v = (z < NG) ? Q(z) : (2.0f * Q(127) - Q(126));
  }
  dst[li] = v;   // regular-temporal: re-read from L2 by the interp kernel
}

// ---- fast path: gather from padded grids; 64 loads = 1 base addr + imm offsets ----
__global__ __launch_bounds__(256) void mg_interp_padded(
    const float* __restrict__ r, const float* __restrict__ phi,
    const float* __restrict__ pm, const float* __restrict__ pp,
    float* __restrict__ out, int n)
{
  int t = blockIdx.x * 256 + threadIdx.x;
  if (t >= n) return;
  // point stream is touched exactly once -> non-temporal, keep L2 for the grids
  float xp = __builtin_nontemporal_load(r + 3 * t + 0);
  float yp = __builtin_nontemporal_load(r + 3 * t + 1);
  float zp = __builtin_nontemporal_load(r + 3 * t + 2);
  float ph = __builtin_nontemporal_load(phi + t);
  Idx ax = axis_idx(xp);
  Idx ay = axis_idx(yp);
  Idx az = axis_idx(zp);
  const float* g  = (ph >= 0.0f) ? pp : pm;       // gather only the selected field
  const float* gb = g + ((ax.i - 2) * NP + (ay.i - 2)) * NP + (az.i - 2);
  float nb[4][4][4];
#pragma unroll
  for (int a = 0; a < 4; ++a)
#pragma unroll
    for (int b = 0; b < 4; ++b)
#pragma unroll
      for (int c = 0; c < 4; ++c)
        nb[a][b][c] = gb[(a * NP + b) * NP + c];  // rows vectorize; all in flight
  __builtin_nontemporal_store(mg_eval(nb, ax.d, ay.d, az.d), out + t);
}

// ---- fallback path (workspace too small): clamp indices, fix up ghosts in-regs ----
__global__ __launch_bounds__(256) void mg_interp_direct(
    const float* __restrict__ r, const float* __restrict__ phi,
    const float* __restrict__ gm, const float* __restrict__ gp,
    float* __restrict__ out, int n)
{
  int t = blockIdx.x * 256 + threadIdx.x;
  if (t >= n) return;
  float xp = __builtin_nontemporal_load(r + 3 * t + 0);
  float yp = __builtin_nontemporal_load(r + 3 * t + 1);
  float zp = __builtin_nontemporal_load(r + 3 * t + 2);
  float ph = __builtin_nontemporal_load(phi + t);
  Idx ax = axis_idx(xp);
  Idx ay = axis_idx(yp);
  Idx az = axis_idx(zp);
  const float* g = (ph >= 0.0f) ? gp : gm;
  int xo[4], yo[4];
#pragma unroll
  for (int s = 0; s < 4; ++s) {
    int mx = ax.i - 2 + s; xo[s] = (mx > 127 ? 127 : mx) << 14;
    int my = ay.i - 2 + s; yo[s] = (my > 127 ? 127 : my) << 7;
  }
  int zb = az.i - 2;                 // zb+2 <= 127 always
  int z3 = (az.i == 127) ? 2 : 3;    // clamped 4th z-slot
  float nb[4][4][4];
#pragma unroll
  for (int a = 0; a < 4; ++a)
#pragma unroll
    for (int b = 0; b < 4; ++b) {
      const float* row = g + (xo[a] + yo[b] + zb);
      nb[a][b][0] = row[0];          // constant offsets -> b96-vectorizable
      nb[a][b][1] = row[1];
      nb[a][b][2] = row[2];
      nb[a][b][3] = row[z3];
    }
  // Ghost fix-ups in reference order x, then y, then z; each later fix-up
  // consumes already-fixed values, reproducing the sequential extrapolation.
  if (ax.i == 127) {
#pragma unroll
    for (int b = 0; b < 4; ++b)
#pragma unroll
      for (int c = 0; c < 4; ++c)
        nb[3][b][c] = 2.0f * nb[2][b][c] - nb[1][b][c];
  }
  if (ay.i == 127) {
#pragma unroll
    for (int a = 0; a < 4; ++a)
#pragma unroll
      for (int c = 0; c < 4; ++c)
        nb[a][3][c] = 2.0f * nb[a][2][c] - nb[a][1][c];
  }
  if (az.i == 127) {
#pragma unroll
    for (int a = 0; a < 4; ++a)
#pragma unroll
      for (int b = 0; b < 4; ++b)
        nb[a][b][3] = 2.0f * nb[a][b][2] - nb[a][b][1];
  }
  __builtin_nontemporal_store(mg_eval(nb, ax.d, ay.d, az.d), out + t);
}

} // namespace

extern "C" void kernel_launch(void* const* d_in, const int* in_sizes, int n_in,
                              void* d_out, int out_size, void* d_ws, size_t ws_size,
                              hipStream_t stream) {
  const float* r   = (const float*)d_in[0];   // [n,3]
  const float* phi = (const float*)d_in[1];   // [n]
  const float* gm  = (const float*)d_in[2];   // [128^3]
  const float* gp  = (const float*)d_in[3];   // [128^3]
  float* out = (float*)d_out;
  int n = in_sizes[1];
  if (n <= 0) return;

  const size_t need = (size_t)(2 * PADN) * sizeof(float);
  int blocks = (n + 255) / 256;
  if (ws_size >= need) {
    float* pad = (float*)d_ws;
    dim3 pg((unsigned)((PADN + 255) / 256), 2);
    pad_ghost_kernel<<<pg, 256, 0, stream>>>(gm, gp, pad);
    mg_interp_padded<<<blocks, 256, 0, stream>>>(r, phi, pad, pad + PADN, out, n);
  } else {
    mg_interp_direct<<<blocks, 256, 0, stream>>>(r, phi, gm, gp, out, n);
  }
}